// LocalOperator_35734127902787
// MI455X (gfx1250) — compile-verified
//
#include <hip/hip_runtime.h>
#include <math.h>

#define BB    4
#define TIN   12
#define TOUT  4
#define UD    3
#define CW    64
#define NDEPTH 4
#define XMM   16
#define YMM   16
#define XRES  128
#define YRES  128
#define GXR   64
#define KXT   32           // 2*XMM x-modes kept
#define BT    (BB*TIN)     // 48
#define PLANE (XRES*YRES)  // 16384

static const size_t NX = (size_t)BT * CW * PLANE;   // 50,331,648 floats

typedef float v2f __attribute__((ext_vector_type(2)));
typedef float v8f __attribute__((ext_vector_type(8)));

__device__ __forceinline__ v8f wmma4(v2f a, v2f b, v8f c) {
  // V_WMMA_F32_16X16X4_F32 : D = A(16x4) x B(4x16) + C(16x16)
  return __builtin_amdgcn_wmma_f32_16x16x4_f32(false, a, false, b, (short)0, c, false, false);
}

// --------------------------- async LDS staging ------------------------------
typedef int gv4i __attribute__((vector_size(16)));   // matches builtin proto
typedef int gv2i __attribute__((vector_size(8)));

#if defined(__gfx1250__) && __has_builtin(__builtin_amdgcn_global_load_async_to_lds_b128)
#define HAVE_ASYNC128 1
__device__ __forceinline__ void async_cp128(void* lds_dst, const void* gsrc) {
  __builtin_amdgcn_global_load_async_to_lds_b128(
      (__attribute__((address_space(1))) gv4i*)(gsrc),
      (__attribute__((address_space(3))) gv4i*)(lds_dst), 0, 0);
}
#endif
#if defined(__gfx1250__) && __has_builtin(__builtin_amdgcn_global_load_async_to_lds_b64)
#define HAVE_ASYNC64 1
__device__ __forceinline__ void async_cp64(void* lds_dst, const void* gsrc) {
  __builtin_amdgcn_global_load_async_to_lds_b64(
      (__attribute__((address_space(1))) gv2i*)(gsrc),
      (__attribute__((address_space(3))) gv2i*)(lds_dst), 0, 0);
}
#endif
#if defined(__gfx1250__)
__device__ __forceinline__ void async_wait0() {
#if __has_builtin(__builtin_amdgcn_s_wait_asynccnt)
  __builtin_amdgcn_s_wait_asynccnt(0);
#else
  asm volatile("s_wait_asynccnt 0x0" ::: "memory");
#endif
}
#endif

// ---------------------------------------------------------------------------
// Twiddle tables (truncated DFT matrices, normalization folded into inverse-y)
// layout inside tw:
//   FyC [m][k]   @     0 (2048)   FyS @  2048
//   FxC [n][kx]  @  4096 (4096)   FxS @  8192
//   GxC [n][kx]  @ 12288 (4096)   GxS @ 16384
//   GyC [y][k]   @ 20480 (2048)   GyS @ 22528
__global__ void k_twiddle(float* __restrict__ tw) {
  int t = blockIdx.x * 256 + threadIdx.x;   // 12288 threads
  const float twopi = 6.283185307179586f;
  if (t < 2048) {                       // forward y-DFT: e^{-i 2pi k m /128}
    int m = t >> 4, k = t & 15;
    float th = twopi * (float)(k * m) / 128.f;
    tw[t]        =  cosf(th);
    tw[2048 + t] = -sinf(th);
  } else if (t < 6144) {                // forward x-DFT (kx 0..15 -> 0..15, 16..31 -> 112..127)
    int e = t - 2048; int n = e >> 5, kx = e & 31;
    int p = (kx < 16) ? kx : kx + 96;
    float th = twopi * (float)(p * n) / 128.f;
    tw[4096 + e] =  cosf(th);
    tw[8192 + e] = -sinf(th);
  } else if (t < 10240) {               // inverse x-DFT: e^{+i 2pi p n /128}
    int e = t - 6144; int n = e >> 5, kx = e & 31;
    int p = (kx < 16) ? kx : kx + 96;
    float th = twopi * (float)(p * n) / 128.f;
    tw[12288 + e] = cosf(th);
    tw[16384 + e] = sinf(th);
  } else {                              // inverse y with conj-symmetry weight + 1/(128*128)
    int e = t - 10240; int y = e >> 4, k = e & 15;
    float th = twopi * (float)(k * y) / 128.f;
    float w = ((k == 0) ? 1.f : 2.f) / 16384.f;
    tw[20480 + e] =  w * cosf(th);
    tw[22528 + e] = -w * sinf(th);
  }
}

// ---------------------------------------------------------------------------
// Lifting: x[bt][o][x][y] = P_w[o][:] . input[bt][:][x][y] + P_b[o]
__global__ void k_lift(const float* __restrict__ in, const float* __restrict__ Pw,
                       const float* __restrict__ Pb, float* __restrict__ x) {
  int g = blockIdx.x * 256 + threadIdx.x;   // 786432
  int bt = g >> 14, xy = g & 16383;
  const float* ib = in + (size_t)bt * UD * PLANE + xy;
  float i0 = ib[0], i1 = ib[PLANE], i2 = ib[2 * PLANE];
  float* xb = x + (size_t)bt * CW * PLANE + xy;
#pragma unroll
  for (int o = 0; o < CW; ++o)
    xb[(size_t)o * PLANE] = Pw[o * 3] * i0 + Pw[o * 3 + 1] * i1 + Pw[o * 3 + 2] * i2 + Pb[o];
}

// ---------------------------------------------------------------------------
// Forward y-DFT: Ay[btc][ky][n] (complex) = sum_m x[btc][n][m] * Fy[m][ky]
__global__ void k_dfty(const float* __restrict__ x, const float* __restrict__ tw,
                       float* __restrict__ Ay) {
  __shared__ __align__(16) float rows[16 * 128];
  int tid = threadIdx.x;              // 256
  int r0 = blockIdx.x * 16;           // 16 consecutive rows (same btc)
  const float* src = x + (size_t)r0 * 128;
#ifdef HAVE_ASYNC128
#pragma unroll
  for (int j = 0; j < 2; ++j) {
    int e = tid + 256 * j;            // 512 x 16B chunks
    async_cp128(&rows[e * 4], src + e * 4);
  }
  async_wait0();
#else
#pragma unroll
  for (int j = 0; j < 8; ++j) rows[tid + j * 256] = src[tid + j * 256];
#endif
  __syncthreads();
  int k = tid & 15, r = tid >> 4;
  const float* FyC = tw;
  const float* FyS = tw + 2048;
  float ar = 0.f, ai = 0.f;
#pragma unroll 4
  for (int m = 0; m < 128; ++m) {
    float v = rows[r * 128 + m];
    ar += v * FyC[m * 16 + k];
    ai += v * FyS[m * 16 + k];
  }
  int row = r0 + r;
  int btc = row >> 7, xn = row & 127;
  size_t o = (((size_t)btc * 16 + k) * 128 + xn) * 2;
  Ay[o] = ar; Ay[o + 1] = ai;
}

// ---------------------------------------------------------------------------
// Forward x-DFT: Mf[mode=kx*16+ky][bt][c] = sum_n Fx[kx][n] * Ay[btc][ky][n]
__global__ void k_dftx(const float* __restrict__ Ay, const float* __restrict__ tw,
                       float* __restrict__ Mf) {
  int g = blockIdx.x * 256 + threadIdx.x;   // 1,572,864
  int kx = g & 31, ky = (g >> 5) & 15, c = (g >> 9) & 63, bt = g >> 15;
  const float* FxC = tw + 4096;
  const float* FxS = tw + 8192;
  int btc = bt * 64 + c;
  const float* arow = Ay + (((size_t)btc * 16 + ky) * 128) * 2;
  float mr = 0.f, mi = 0.f;
#pragma unroll 4
  for (int n = 0; n < 128; ++n) {
    float xr = arow[n * 2], xi = arow[n * 2 + 1];       // broadcast across wave
    float fr = FxC[n * 32 + kx], fi = FxS[n * 32 + kx]; // coalesced
    mr += fr * xr - fi * xi;
    mi += fr * xi + fi * xr;
  }
  int mode = kx * 16 + ky;
  size_t o = (((size_t)mode * 48 + bt) * 64 + c) * 2;
  Mf[o] = mr; Mf[o + 1] = mi;
}

// ---------------------------------------------------------------------------
// Per-mode complex matmul with spectral weights (WMMA f32 16x16x4):
//   Mo[mode][bt][o] = sum_i Mf[mode][bt][i] * W[i][o]   (W = w1 or w2 per kx)
// Mf mode-slice staged interleaved (re/im) via async-to-LDS.
__global__ void k_spec(int d, const float* __restrict__ Mf,
                       const float* __restrict__ w1r, const float* __restrict__ w1i,
                       const float* __restrict__ w2r, const float* __restrict__ w2i,
                       float* __restrict__ Mo) {
  __shared__ __align__(16) float Xc[6144];            // interleaved [bt*64+i][ri]
  __shared__ __align__(16) float Wr[4096], Wi[4096];  // 56 KB total
  int tid = threadIdx.x;     // 128 threads = 4 waves
  int mode = blockIdx.x;     // 0..511
  int kx = mode >> 4, ky = mode & 15;
  const float* src = Mf + (size_t)mode * 48 * 64 * 2;
#ifdef HAVE_ASYNC128
#pragma unroll
  for (int j = 0; j < 12; ++j) {
    int e = tid + 128 * j;            // 1536 x 16B chunks
    async_cp128(&Xc[e * 4], src + e * 4);
  }
#else
  for (int e = tid; e < 6144; e += 128) Xc[e] = src[e];
#endif
  const float* wr; const float* wi; int kxl;
  if (kx < 16) { wr = w1r; wi = w1i; kxl = kx; } else { wr = w2r; wi = w2i; kxl = kx - 16; }
  size_t wofs = (size_t)d * 64 * 64 * 256 + (size_t)kxl * 16 + ky;
  for (int e = tid; e < 4096; e += 128) {      // e = i*64 + o, per-element stride 256
    size_t a = wofs + (size_t)e * 256;
    Wr[e] = wr[a]; Wi[e] = wi[a];
  }
#ifdef HAVE_ASYNC128
  async_wait0();
#endif
  __syncthreads();
  int L = tid & 31, wv = tid >> 5;   // wave = o-tile
  int lo = L & 15, hi = L >> 4;
  int ocol = wv * 16 + lo;
  v8f z = {0.f,0.f,0.f,0.f,0.f,0.f,0.f,0.f};
  v8f cr[3] = {z, z, z}, ci[3] = {z, z, z};
#pragma unroll
  for (int ks = 0; ks < 16; ++ks) {
    int i0 = ks * 4 + 2 * hi;
    v2f br  = { Wr[i0 * 64 + ocol], Wr[i0 * 64 + 64 + ocol] };
    v2f bi  = { Wi[i0 * 64 + ocol], Wi[i0 * 64 + 64 + ocol] };
    v2f bin = { -bi.x, -bi.y };     // f32 WMMA has no A/B NEG modifier
#pragma unroll
    for (int m = 0; m < 3; ++m) {
      int e0 = (m * 16 + lo) * 64 + i0;
      v2f ar = { Xc[e0 * 2],     Xc[e0 * 2 + 2] };
      v2f ai = { Xc[e0 * 2 + 1], Xc[e0 * 2 + 3] };
      cr[m] = wmma4(ar, br,  cr[m]);   // +Xr*Wr
      cr[m] = wmma4(ai, bin, cr[m]);   // -Xi*Wi
      ci[m] = wmma4(ar, bi,  ci[m]);   // +Xr*Wi
      ci[m] = wmma4(ai, br,  ci[m]);   // +Xi*Wr
    }
  }
#pragma unroll
  for (int m = 0; m < 3; ++m) {
#pragma unroll
    for (int v = 0; v < 8; ++v) {
      int btr = m * 16 + v + 8 * hi;
      size_t o = (((size_t)mode * 48 + btr) * 64 + ocol) * 2;
      Mo[o] = cr[m][v]; Mo[o + 1] = ci[m][v];
    }
  }
}

// ---------------------------------------------------------------------------
// Inverse x-DFT: Ax[bt][c][x][ky] (complex) = sum_kx Gx[x][kx] * Mo[kx*16+ky][bt][c]
__global__ void k_invx(const float* __restrict__ Mo, const float* __restrict__ tw,
                       float* __restrict__ Ax) {
  __shared__ __align__(16) float Ml[1024];
  int tid = threadIdx.x;               // 256
  int bt = blockIdx.x >> 6, c = blockIdx.x & 63;
#ifdef HAVE_ASYNC64
#pragma unroll
  for (int j = 0; j < 2; ++j) {
    int mode = tid + 256 * j;          // 512 x 8B chunks (one complex pair each)
    async_cp64(&Ml[mode * 2], Mo + (((size_t)mode * 48 + bt) * 64 + c) * 2);
  }
  async_wait0();
#else
#pragma unroll
  for (int j = 0; j < 4; ++j) {
    int e = tid * 4 + j;               // 0..1023
    int mode = e >> 1, ri = e & 1;
    Ml[e] = Mo[(((size_t)mode * 48 + bt) * 64 + c) * 2 + ri];
  }
#endif
  __syncthreads();
  const float* GxC = tw + 12288;
  const float* GxS = tw + 16384;
#pragma unroll
  for (int j = 0; j < 8; ++j) {
    int p = tid + 256 * j;             // 0..2047
    int ky = p & 15, n = p >> 4;
    float ar = 0.f, ai = 0.f;
#pragma unroll 8
    for (int kx = 0; kx < 32; ++kx) {
      float mr = Ml[(kx * 16 + ky) * 2], mi = Ml[(kx * 16 + ky) * 2 + 1];
      float gr = GxC[n * 32 + kx], gi = GxS[n * 32 + kx];
      ar += gr * mr - gi * mi;
      ai += gr * mi + gi * mr;
    }
    size_t o = ((size_t)(bt * 64 + c) * 128 + n) * 32 + ky * 2;
    Ax[o] = ar; Ax[o + 1] = ai;
  }
}

// ---------------------------------------------------------------------------
// Fused epilogue per (bt, x): WMMA(inverse-y DFT K=32 + 1x1 channel mix K=64)
// -> +bias -> LayerNorm(channels) -> GELU -> +bilinear context -> store.
// C tiles: M = channels (4 tiles), N = 16 y-points (wave = y-tile).
__global__ void k_fuse(int d, const float* __restrict__ x, const float* __restrict__ Ax,
                       const float* __restrict__ tw,
                       const float* __restrict__ llw, const float* __restrict__ llb,
                       const float* __restrict__ lng, const float* __restrict__ lnb,
                       const float* __restrict__ gc, float* __restrict__ xnext) {
  __shared__ __align__(16) float Xl[64 * 128];   // x[i][y] for this (bt, xrow)   32 KB
  __shared__ __align__(16) float Wl[4096];       // ll_w[d][o][i]                 16 KB
  __shared__ __align__(16) float Ais[2048];      // Ax slice [o][ky][ri]           8 KB
  int tid = threadIdx.x;           // 256 = 8 waves
  int bt = blockIdx.x >> 7, xrow = blockIdx.x & 127;
  int b = bt / TIN, t = bt % TIN;

  const float* xb = x + (size_t)bt * CW * PLANE + (size_t)xrow * 128;
#ifdef HAVE_ASYNC128
#pragma unroll
  for (int j = 0; j < 8; ++j) {
    int e = tid + 256 * j;            // 2048 x 16B chunks of Xl
    async_cp128(&Xl[e * 4], xb + (size_t)(e >> 5) * PLANE + (e & 31) * 4);
  }
#pragma unroll
  for (int j = 0; j < 4; ++j) {
    int e = tid + 256 * j;            // 1024 x 16B chunks of Wl
    async_cp128(&Wl[e * 4], llw + (size_t)d * 4096 + e * 4);
  }
#pragma unroll
  for (int j = 0; j < 2; ++j) {
    int e = tid + 256 * j;            // 512 x 16B chunks of Ais
    async_cp128(&Ais[e * 4],
                Ax + ((size_t)(bt * 64 + (e >> 3)) * 128 + xrow) * 32 + (e & 7) * 4);
  }
  async_wait0();
#else
#pragma unroll
  for (int j = 0; j < 32; ++j) {
    int e = tid + 256 * j;
    Xl[e] = xb[(size_t)(e >> 7) * PLANE + (e & 127)];
  }
#pragma unroll
  for (int j = 0; j < 16; ++j) {
    int e = tid + 256 * j;
    Wl[e] = llw[(size_t)d * 4096 + e];
  }
#pragma unroll
  for (int j = 0; j < 8; ++j) {
    int e = tid + 256 * j;           // e = o*32 + q
    Ais[e] = Ax[((size_t)(bt * 64 + (e >> 5)) * 128 + xrow) * 32 + (e & 31)];
  }
#endif
  __syncthreads();

  int L = tid & 31, wv = tid >> 5;   // wave = y-tile
  int lo = L & 15, hi = L >> 4;
  int y = wv * 16 + lo;
  const float* GyC = tw + 20480;
  const float* GyS = tw + 22528;

  v8f z = {0.f,0.f,0.f,0.f,0.f,0.f,0.f,0.f};
  v8f acc[4] = {z, z, z, z};

  // 1x1 channel mix: out[o][y] += ll_w[o][i] * x[i][y]   (K = 64)
#pragma unroll
  for (int ks = 0; ks < 16; ++ks) {
    int i0 = ks * 4 + 2 * hi;
    v2f bvec = { Xl[i0 * 128 + y], Xl[(i0 + 1) * 128 + y] };
#pragma unroll
    for (int m = 0; m < 4; ++m) {
      int o = m * 16 + lo;
      v2f a = { Wl[o * 64 + i0], Wl[o * 64 + i0 + 1] };
      acc[m] = wmma4(a, bvec, acc[m]);
    }
  }
  // inverse-y DFT: out[o][y] += Re{Ax[o][ky]} * GyC[y][ky] + Im{...} * GyS[y][ky]  (K = 32)
#pragma unroll
  for (int ks = 0; ks < 8; ++ks) {
    int kyA = ks * 2 + hi;
    v2f bvec = { GyC[y * 16 + kyA], GyS[y * 16 + kyA] };
#pragma unroll
    for (int m = 0; m < 4; ++m) {
      int o = m * 16 + lo;
      v2f a = { Ais[o * 32 + kyA * 2], Ais[o * 32 + kyA * 2 + 1] };
      acc[m] = wmma4(a, bvec, acc[m]);
    }
  }

  // LayerNorm over 64 channels: lanes L and L^16 jointly hold all channels of point y
  float vals[32];
  float s1 = 0.f, s2 = 0.f;
#pragma unroll
  for (int m = 0; m < 4; ++m) {
#pragma unroll
    for (int v = 0; v < 8; ++v) {
      int o = m * 16 + v + 8 * hi;
      float val = acc[m][v] + llb[d * 64 + o];
      vals[m * 8 + v] = val;
      s1 += val; s2 += val * val;
    }
  }
  s1 += __shfl_xor(s1, 16, 32);
  s2 += __shfl_xor(s2, 16, 32);
  float mu   = s1 * (1.f / 64.f);
  float var  = s2 * (1.f / 64.f) - mu * mu;
  float rinv = rsqrtf(var + 1e-5f);

  // bilinear context coords (64 -> 128, half-pixel centers, edge clamp)
  float sx = 0.5f * (float)xrow - 0.25f;
  int ixf = (int)floorf(sx);
  float wx = sx - (float)ixf;
  int ix0 = min(max(ixf, 0), 63), ix1 = min(max(ixf + 1, 0), 63);
  float sy = 0.5f * (float)y - 0.25f;
  int iyf = (int)floorf(sy);
  float wy = sy - (float)iyf;
  int iy0 = min(max(iyf, 0), 63), iy1 = min(max(iyf + 1, 0), 63);
  const float* gcb = gc + ((((size_t)d * BB + b) * TIN + t) * CW) * 4096;

#pragma unroll
  for (int m = 0; m < 4; ++m) {
#pragma unroll
    for (int v = 0; v < 8; ++v) {
      int o = m * 16 + v + 8 * hi;
      float nv = (vals[m * 8 + v] - mu) * rinv * lng[d * 64 + o] + lnb[d * 64 + o];
      float ge = 0.5f * nv * (1.f + erff(nv * 0.70710678118f));   // exact GELU
      const float* cp = gcb + (size_t)o * 4096;
      float g00 = cp[ix0 * 64 + iy0], g01 = cp[ix0 * 64 + iy1];
      float g10 = cp[ix1 * 64 + iy0], g11 = cp[ix1 * 64 + iy1];
      float gl = (1.f - wx) * ((1.f - wy) * g00 + wy * g01)
               +        wx  * ((1.f - wy) * g10 + wy * g11);
      xnext[((size_t)(bt * 64 + o) * 128 + xrow) * 128 + y] = ge + gl;
    }
  }
}

// ---------------------------------------------------------------------------
// Temporal aggregation (12->4) + projection (64->3), fused; single pass over x.
__global__ void k_final(const float* __restrict__ x, const float* __restrict__ Wtw,
                        const float* __restrict__ Wtb, const float* __restrict__ Qw,
                        const float* __restrict__ Qb, float* __restrict__ out) {
  int g = blockIdx.x * 256 + threadIdx.x;   // 65536
  int b = g >> 14, xy = g & 16383;
  float acc[TOUT][UD] = {};
  const float* xb = x + (size_t)b * TIN * CW * PLANE + xy;
  for (int c = 0; c < CW; ++c) {
    float tmp[TOUT];
#pragma unroll
    for (int to = 0; to < TOUT; ++to) tmp[to] = Wtb[to];
#pragma unroll
    for (int t = 0; t < TIN; ++t) {
      float v = xb[((size_t)t * CW + c) * PLANE];
#pragma unroll
      for (int to = 0; to < TOUT; ++to) tmp[to] += Wtw[to * TIN + t] * v;
    }
#pragma unroll
    for (int u = 0; u < UD; ++u) {
      float q = Qw[u * CW + c];
#pragma unroll
      for (int to = 0; to < TOUT; ++to) acc[to][u] += q * tmp[to];
    }
  }
#pragma unroll
  for (int to = 0; to < TOUT; ++to)
#pragma unroll
    for (int u = 0; u < UD; ++u)
      out[(((size_t)b * TOUT + to) * UD + u) * PLANE + xy] = acc[to][u] + Qb[u];
}

// ---------------------------------------------------------------------------
extern "C" void kernel_launch(void* const* d_in, const int* in_sizes, int n_in,
                              void* d_out, int out_size, void* d_ws, size_t ws_size,
                              hipStream_t stream) {
  const float* input = (const float*)d_in[0];
  const float* gc    = (const float*)d_in[1];
  const float* Pw    = (const float*)d_in[2];
  const float* Pb    = (const float*)d_in[3];
  const float* Qw    = (const float*)d_in[4];
  const float* Qb    = (const float*)d_in[5];
  const float* Wtw   = (const float*)d_in[6];
  const float* Wtb   = (const float*)d_in[7];
  const float* w1r   = (const float*)d_in[8];
  const float* w1i   = (const float*)d_in[9];
  const float* w2r   = (const float*)d_in[10];
  const float* w2i   = (const float*)d_in[11];
  const float* llw   = (const float*)d_in[12];
  const float* llb   = (const float*)d_in[13];
  const float* lng   = (const float*)d_in[14];
  const float* lnb   = (const float*)d_in[15];

  float* ws = (float*)d_ws;
  float* x0 = ws;                       // 50,331,648
  float* x1 = ws + NX;                  // 50,331,648
  float* Ab = ws + 2 * NX;              // 12,582,912 (forward-y / inverse-x scratch)
  float* Mf = Ab + 12582912;            //  3,145,728
  float* Mo = Mf + 3145728;             //  3,145,728
  float* tw = Mo + 3145728;             //     24,576

  k_twiddle<<<48, 256, 0, stream>>>(tw);
  k_lift<<<3072, 256, 0, stream>>>(input, Pw, Pb, x0);

  float* xc = x0;
  float* xn = x1;
  for (int d = 0; d < NDEPTH; ++d) {
    k_dfty<<<24576, 256, 0, stream>>>(xc, tw, Ab);
    k_dftx<<<6144, 256, 0, stream>>>(Ab, tw, Mf);
    k_spec<<<512, 128, 0, stream>>>(d, Mf, w1r, w1i, w2r, w2i, Mo);
    k_invx<<<3072, 256, 0, stream>>>(Mo, tw, Ab);
    k_fuse<<<6144, 256, 0, stream>>>(d, xc, Ab, tw, llw, llb, lng, lnb, gc, xn);
    float* tswap = xc; xc = xn; xn = tswap;
  }
  k_final<<<256, 256, 0, stream>>>(xc, Wtw, Wtb, Qw, Qb, (float*)d_out);
}